// DenseCaps_63591285785045
// MI455X (gfx1250) — compile-verified
//
#include <hip/hip_runtime.h>
#include <math.h>

// ---------------------------------------------------------------------------
// DenseCaps dynamic routing, MI455X (gfx1250, wave32).
// Strategy: never materialize preds (268MB > 192MB L2). Recompute per pass
// from x (4MB) + W (32MB), which stay L2-resident. All matrix work goes via
// V_WMMA_F32_16X16X4_F32: M=16 (D_out), N=16 (batch tile), K=4 (2 ops = D_in=8).
// ---------------------------------------------------------------------------

#define B_SZ   64
#define N_IN   2048
#define D_IN   8
#define N_OUT  32
#define D_OUT  16
#define EPSV   1e-7f

#define CHUNK_C 16                    // n per wave, route kernel (512 waves)
#define CHUNK_D 128                   // n per wave, predsum kernel
#define NCHUNK_D (N_IN / CHUNK_D)     // 16 partial buffers

typedef __attribute__((ext_vector_type(2))) float v2f;
typedef __attribute__((ext_vector_type(8))) float v8f;

__device__ __forceinline__ v8f wmma4(v2f a, v2f b, v8f c) {
    // (neg_a, A, neg_b, B, c_mod, C, reuse_a, reuse_b)
    return __builtin_amdgcn_wmma_f32_16x16x4_f32(false, a, false, b,
                                                 (short)0, c, false, false);
}

// s_partial[chunk][b,j,d] = sum_{n in chunk} rw[b,n,j] * (W[n,j] @ x[b,n])
// One wave per (j, btile, chunk). USE_RW=false => uniform 1/32 (iteration 0).
template <bool USE_RW>
__global__ void __launch_bounds__(128)
predsum_kernel(const float* __restrict__ x, const float* __restrict__ W,
               const float* __restrict__ rw, float* __restrict__ part) {
    const int lane  = threadIdx.x & 31;
    const int wid   = blockIdx.x * (blockDim.x >> 5) + (threadIdx.x >> 5);
    const int j     = wid & (N_OUT - 1);
    const int btile = (wid >> 5) & 3;
    const int chunk = wid >> 7;
    const int hi    = lane >> 4;          // 0: lanes 0-15, 1: lanes 16-31
    const int b     = btile * 16 + (lane & 15);
    const int d_lo  = lane & 15;          // A-matrix row (M)

    v8f acc = {};
    const int n0 = chunk * CHUNK_D;
    for (int n = n0; n < n0 + CHUNK_D; ++n) {
        // B fragments: x[b,n,0..7] column, scaled by routing weight
        const float* xr = x + ((size_t)b * N_IN + n) * D_IN;
        float4 lo = *(const float4*)xr;
        float4 hp = *(const float4*)(xr + 4);
        float w = USE_RW ? rw[((size_t)b * N_IN + n) * N_OUT + j] : 0.03125f;
        v2f b0, b1;
        b0[0] = (hi ? lo.y : lo.x) * w;   // K = 0/1
        b0[1] = (hi ? lo.w : lo.z) * w;   // K = 2/3
        b1[0] = (hi ? hp.y : hp.x) * w;   // K = 4/5
        b1[1] = (hi ? hp.w : hp.z) * w;   // K = 6/7
        // A fragments: W[n,j,d,0..7]
        const float* wb = W + (((size_t)n * N_OUT + j) * D_OUT + d_lo) * D_IN + hi * 2;
        v2f a0 = *(const v2f*)wb;
        v2f a1 = *(const v2f*)(wb + 4);
        // prefetch the A-fragment stream a few n ahead (global_prefetch_b8)
        __builtin_prefetch(wb + 8 * (size_t)N_OUT * D_OUT * D_IN, 0, 3);
        acc = wmma4(a0, b0, acc);
        acc = wmma4(a1, b1, acc);
    }
    // C/D layout: VGPR r -> d = r + 8*hi; each lane owns 8 consecutive d.
    float* p = part + (size_t)chunk * (B_SZ * N_OUT * D_OUT)
                    + ((size_t)b * N_OUT + j) * D_OUT + hi * 8;
    *(float4*)p       = make_float4(acc[0], acc[1], acc[2], acc[3]);
    *(float4*)(p + 4) = make_float4(acc[4], acc[5], acc[6], acc[7]);
}

// Fixed-order reduce of chunk partials + squash.  One thread per (b,j).
__global__ void __launch_bounds__(256)
squash_kernel(const float* __restrict__ part, float* __restrict__ vout) {
    int t = blockIdx.x * blockDim.x + threadIdx.x;
    if (t >= B_SZ * N_OUT) return;
    float s[D_OUT];
#pragma unroll
    for (int d = 0; d < D_OUT; ++d) s[d] = 0.f;
    for (int c = 0; c < NCHUNK_D; ++c) {
        const float* p = part + (size_t)c * (B_SZ * N_OUT * D_OUT) + (size_t)t * D_OUT;
#pragma unroll
        for (int q = 0; q < 4; ++q) {
            float4 r = *(const float4*)(p + q * 4);
            s[q*4+0] += r.x; s[q*4+1] += r.y; s[q*4+2] += r.z; s[q*4+3] += r.w;
        }
    }
    float s2 = 0.f;
#pragma unroll
    for (int d = 0; d < D_OUT; ++d) s2 += s[d] * s[d];
    float scale = s2 / (1.0f + s2) / sqrtf(s2 + EPSV);
    float* o = vout + (size_t)t * D_OUT;
#pragma unroll
    for (int q = 0; q < 4; ++q)
        *(float4*)(o + q * 4) = make_float4(s[q*4+0]*scale, s[q*4+1]*scale,
                                            s[q*4+2]*scale, s[q*4+3]*scale);
}

// Agreement + raw update + softmax.  One wave per (btile, n-chunk).
// HAS_RAW_IN=false => previous raw was zero.  WRITE_RAW => store updated raw.
template <bool HAS_RAW_IN, bool WRITE_RAW>
__global__ void __launch_bounds__(128)
route_kernel(const float* __restrict__ x, const float* __restrict__ W,
             const float* __restrict__ v,
             const float* __restrict__ raw_in, float* __restrict__ raw_out,
             float* __restrict__ rw_out) {
    const int lane  = threadIdx.x & 31;
    const int wid   = blockIdx.x * (blockDim.x >> 5) + (threadIdx.x >> 5);
    const int btile = wid & 3;
    const int chunk = wid >> 2;
    const int hi    = lane >> 4;
    const int b     = btile * 16 + (lane & 15);
    const int d_lo  = lane & 15;

    const int n0 = chunk * CHUNK_C;
    for (int n = n0; n < n0 + CHUNK_C; ++n) {
        // B fragments (shared by all 32 output capsules)
        const float* xr = x + ((size_t)b * N_IN + n) * D_IN;
        float4 lo = *(const float4*)xr;
        float4 hp = *(const float4*)(xr + 4);
        v2f b0, b1;
        b0[0] = hi ? lo.y : lo.x;  b0[1] = hi ? lo.w : lo.z;
        b1[0] = hi ? hp.y : hp.x;  b1[1] = hi ? hp.w : hp.z;

        float a[N_OUT];
#pragma unroll
        for (int j = 0; j < N_OUT; ++j) {
            const float* wb = W + (((size_t)n * N_OUT + j) * D_OUT + d_lo) * D_IN + hi * 2;
            v2f a0 = *(const v2f*)wb;
            v2f a1 = *(const v2f*)(wb + 4);
            // prefetch next n's fragment for this j (global_prefetch_b8)
            __builtin_prefetch(wb + (size_t)N_OUT * D_OUT * D_IN, 0, 3);
            v8f p = {};
            p = wmma4(a0, b0, p);
            p = wmma4(a1, b1, p);
            // agreement: sum_d preds[d,b] * v[b,j,d]; this lane holds d = hi*8 .. hi*8+7
            const float* vp = v + ((size_t)b * N_OUT + j) * D_OUT + hi * 8;
            float4 va = *(const float4*)vp;
            float4 vb = *(const float4*)(vp + 4);
            float dot = p[0]*va.x + p[1]*va.y + p[2]*va.z + p[3]*va.w
                      + p[4]*vb.x + p[5]*vb.y + p[6]*vb.z + p[7]*vb.w;
            dot += __shfl_xor(dot, 16, 32);     // combine d-halves (wave32)
            a[j] = dot;
        }
        const size_t rbase = ((size_t)b * N_IN + n) * N_OUT;
        if (HAS_RAW_IN) {
#pragma unroll
            for (int q = 0; q < 8; ++q) {
                float4 r = *(const float4*)(raw_in + rbase + q * 4);
                a[q*4+0] += r.x; a[q*4+1] += r.y; a[q*4+2] += r.z; a[q*4+3] += r.w;
            }
        }
        if (WRITE_RAW && lane < 16) {
#pragma unroll
            for (int q = 0; q < 8; ++q)
                *(float4*)(raw_out + rbase + q * 4) =
                    make_float4(a[q*4+0], a[q*4+1], a[q*4+2], a[q*4+3]);
        }
        // softmax over j (32 registers)
        float m = a[0];
#pragma unroll
        for (int j = 1; j < N_OUT; ++j) m = fmaxf(m, a[j]);
        float ssum = 0.f;
#pragma unroll
        for (int j = 0; j < N_OUT; ++j) { a[j] = __expf(a[j] - m); ssum += a[j]; }
        float inv = 1.0f / ssum;
        if (lane < 16) {
#pragma unroll
            for (int q = 0; q < 8; ++q)
                *(float4*)(rw_out + rbase + q * 4) =
                    make_float4(a[q*4+0]*inv, a[q*4+1]*inv, a[q*4+2]*inv, a[q*4+3]*inv);
        }
    }
}

extern "C" void kernel_launch(void* const* d_in, const int* in_sizes, int n_in,
                              void* d_out, int out_size, void* d_ws, size_t ws_size,
                              hipStream_t stream) {
    const float* x = (const float*)d_in[0];   // [64, 2048, 8]
    const float* W = (const float*)d_in[1];   // [2048, 32, 16, 8]
    float* out    = (float*)d_out;
    float* v_out  = out;                          // [64,32,16]
    float* rw_out = out + B_SZ * N_OUT * D_OUT;   // [64,2048,32]

    char*  ws   = (char*)d_ws;
    float* raw  = (float*)ws;                             // 16 MB
    float* rw1  = (float*)(ws + (size_t)16777216);        // 16 MB
    float* vbuf = (float*)(ws + (size_t)2 * 16777216);    // 128 KB
    float* part = (float*)(ws + (size_t)2 * 16777216 + 131072); // 2 MB

    dim3 blk(128);
    dim3 grdD((N_OUT * 4 * NCHUNK_D) / 4);      // 2048 waves -> 512 blocks
    dim3 grdC((4 * (N_IN / CHUNK_C)) / 4);      // 512 waves  -> 128 blocks
    dim3 blkS(256), grdS((B_SZ * N_OUT + 255) / 256);

    // iteration 0: rw uniform = 1/32
    predsum_kernel<false><<<grdD, blk, 0, stream>>>(x, W, nullptr, part);
    squash_kernel<<<grdS, blkS, 0, stream>>>(part, vbuf);
    // iteration 1: raw_1 = agreement(v0); rw_1 = softmax(raw_1)
    route_kernel<false, true><<<grdC, blk, 0, stream>>>(x, W, vbuf, nullptr, raw, rw1);
    predsum_kernel<true><<<grdD, blk, 0, stream>>>(x, W, rw1, part);
    squash_kernel<<<grdS, blkS, 0, stream>>>(part, vbuf);
    // iteration 2: raw_2 = raw_1 + agreement(v1); rw_2 -> output
    route_kernel<true, false><<<grdC, blk, 0, stream>>>(x, W, vbuf, raw, nullptr, rw_out);
    predsum_kernel<true><<<grdD, blk, 0, stream>>>(x, W, rw_out, part);
    squash_kernel<<<grdS, blkS, 0, stream>>>(part, v_out);
}